// Attention_86732569575984
// MI455X (gfx1250) — compile-verified
//
#include <hip/hip_runtime.h>
#include <hip/hip_bf16.h>
#include <math.h>

// Problem constants (reference: B=2, L=2048, DIM=1024, H=16, HD=64)
#define B_   2
#define L_   2048
#define DIM_ 1024
#define H_   16
#define HD_  64

typedef float v2f __attribute__((ext_vector_type(2)));
typedef float v8f __attribute__((ext_vector_type(8)));

// D(16x16,f32) = A(16x4,f32) x B(4x16,f32) + C   -- fp32-exact WMMA (CDNA5)
__device__ __forceinline__ v8f wmma_f32(v2f a, v2f b, v8f c) {
  return __builtin_amdgcn_wmma_f32_16x16x4_f32(false, a, false, b, (short)0, c,
                                               false, false);
}

// CDNA5 async copy: 16B global -> LDS, tracked by ASYNCcnt (no VGPR roundtrip).
// LDS offset = low 32 bits of the generic pointer (flat addr[31:0] is the
// wave-relative LDS offset per the aperture mapping).
__device__ __forceinline__ void async_copy_b128(void* lds_dst,
                                                const float* gsrc) {
  const unsigned lds_off = (unsigned)(unsigned long long)lds_dst;
  asm volatile("global_load_async_to_lds_b128 %0, %1, off"
               :
               : "v"(lds_off), "v"(gsrc)
               : "memory");
}

__device__ __forceinline__ void wait_asynccnt0() {
  asm volatile("s_wait_asynccnt 0x0" ::: "memory");
}

// ---------------------------------------------------------------------------
// GEMM: C[M,N] = A[M,K] * B[N,K]^T   (both row-major, inner dim K)
// Block: 256 threads (8 waves), 128x64 tile, K-step 32.
// Double-buffered LDS staged with global_load_async_to_lds_b128; the async
// engine fills buffer i+1 while WMMAs consume buffer i.
// Wave (wm,wn) in 4x2 grid owns a 32x32 region = 2x2 WMMA tiles.
// Requires M%128==0, N%64==0, K%32==0 (true for all uses here).
// ---------------------------------------------------------------------------
__global__ __launch_bounds__(256) void gemm_f32_nt(const float* __restrict__ A,
                                                   const float* __restrict__ Bm,
                                                   float* __restrict__ C,
                                                   int M, int N, int K) {
  constexpr int BM = 128, BN = 64, BK = 32;
  constexpr int LP = BK + 4;  // pitch 36 floats = 144B (16B-aligned rows)
  __shared__ float As[2][BM * LP];
  __shared__ float Bs[2][BN * LP];
  const int tid  = threadIdx.x;
  const int wave = tid >> 5;
  const int lane = tid & 31;
  const int half = lane >> 4;
  const int ln   = lane & 15;
  const int wm   = wave >> 1;
  const int wn   = wave & 1;
  const int bm   = blockIdx.y * BM;
  const int bn   = blockIdx.x * BN;

  const int q4 = tid & 7;   // which float4 within a 32-float row
  const int r0 = tid >> 3;  // 0..31

  // Issue the async loads for one K-step into LDS buffer `buf`.
  auto stage = [&](int buf, int k0) {
#pragma unroll
    for (int i = 0; i < 4; ++i) {
      const int r = r0 + i * 32;
      async_copy_b128(&As[buf][r * LP + q4 * 4],
                      A + (size_t)(bm + r) * K + k0 + q4 * 4);
    }
#pragma unroll
    for (int i = 0; i < 2; ++i) {
      const int r = r0 + i * 32;
      async_copy_b128(&Bs[buf][r * LP + q4 * 4],
                      Bm + (size_t)(bn + r) * K + k0 + q4 * 4);
    }
  };

  v8f acc[2][2] = {};
  const int steps = K / BK;
  stage(0, 0);

  for (int i = 0; i < steps; ++i) {
    const int cur = i & 1;
    wait_asynccnt0();   // this wave's loads into `cur` have landed in LDS
    __syncthreads();    // everyone's loads landed; everyone done reading cur^1
    if (i + 1 < steps) stage(cur ^ 1, (i + 1) * BK);

    const float* Asb = As[cur];
    const float* Bsb = Bs[cur];
    // A frag (16x4): lane -> M=ln, K=4*kf+2*half+{0,1}; B frag (4x16) mirrored.
#pragma unroll
    for (int kf = 0; kf < BK / 4; ++kf) {
      const int kc = kf * 4 + 2 * half;
      v2f a0 = *(const v2f*)&Asb[(wm * 32 + ln) * LP + kc];
      v2f a1 = *(const v2f*)&Asb[(wm * 32 + 16 + ln) * LP + kc];
      v2f b0 = *(const v2f*)&Bsb[(wn * 32 + ln) * LP + kc];
      v2f b1 = *(const v2f*)&Bsb[(wn * 32 + 16 + ln) * LP + kc];
      acc[0][0] = wmma_f32(a0, b0, acc[0][0]);
      acc[0][1] = wmma_f32(a0, b1, acc[0][1]);
      acc[1][0] = wmma_f32(a1, b0, acc[1][0]);
      acc[1][1] = wmma_f32(a1, b1, acc[1][1]);
    }
  }

  // C layout: VGPR r, lane -> (M = r + 8*half, N = ln)
#pragma unroll
  for (int i = 0; i < 2; ++i)
#pragma unroll
    for (int j = 0; j < 2; ++j)
#pragma unroll
      for (int r = 0; r < 8; ++r) {
        const int row = bm + wm * 32 + i * 16 + r + 8 * half;
        const int col = bn + wn * 32 + j * 16 + ln;
        C[(size_t)row * N + col] = acc[i][j][r];
      }
}

// ---------------------------------------------------------------------------
// Per-(b,h,l) row epilogue: v-blend, RoPE(q,k), k*=ratio, RMS-norm(q,k).
// One wave per row; lane owns elements hd=lane and hd=lane+32 (RoPE pair).
// ---------------------------------------------------------------------------
__global__ __launch_bounds__(256) void qkv_epilogue(
    const float* __restrict__ qkv, const float* __restrict__ v0,
    const float* __restrict__ rc, const float* __restrict__ rsn,
    const float* __restrict__ lamp, float* __restrict__ qw,
    float* __restrict__ kw, float* __restrict__ vw, float* __restrict__ vout) {
  const int wave = threadIdx.x >> 5;
  const int lane = threadIdx.x & 31;
  const int row  = blockIdx.x * 8 + wave;  // linear over (b,h,l)
  const int b    = row >> 15;              // / (H_*L_)
  const int rem  = row & 32767;
  const int h    = rem >> 11;              // / L_
  const int l    = rem & (L_ - 1);

  const float lam   = lamp[0];
  const float ratio = sqrtf(logf((float)L_) / logf(1040.0f));

  const size_t qb = (size_t)(b * L_ + l) * (3 * DIM_) + h * HD_;
  const float q1 = qkv[qb + lane];
  const float q2 = qkv[qb + lane + 32];
  const float k1 = qkv[qb + DIM_ + lane];
  const float k2 = qkv[qb + DIM_ + lane + 32];
  const float x1 = qkv[qb + 2 * DIM_ + lane];
  const float x2 = qkv[qb + 2 * DIM_ + lane + 32];

  const float c = rc[l * (HD_ / 2) + lane];
  const float s = rsn[l * (HD_ / 2) + lane];

  float qy1 = q1 * c + q2 * s;
  float qy2 = -q1 * s + q2 * c;
  float ky1 = (k1 * c + k2 * s) * ratio;
  float ky2 = (-k1 * s + k2 * c) * ratio;

  float qs = qy1 * qy1 + qy2 * qy2;
  float ks = ky1 * ky1 + ky2 * ky2;
#pragma unroll
  for (int off = 16; off > 0; off >>= 1) {
    qs += __shfl_xor(qs, off, 32);
    ks += __shfl_xor(ks, off, 32);
  }
  const float qn = rsqrtf(qs * (1.0f / HD_) + 1e-6f);
  const float kn = rsqrtf(ks * (1.0f / HD_) + 1e-6f);
  qy1 *= qn; qy2 *= qn; ky1 *= kn; ky2 *= kn;

  const size_t ob = (size_t)row * HD_;
  const float w1 = v0[ob + lane];
  const float w2 = v0[ob + lane + 32];
  const float vb1 = lam * x1 + (1.0f - lam) * w1;
  const float vb2 = lam * x2 + (1.0f - lam) * w2;

  qw[ob + lane] = qy1;      qw[ob + lane + 32] = qy2;
  kw[ob + lane] = ky1;      kw[ob + lane + 32] = ky2;
  vw[ob + lane] = vb1;      vw[ob + lane + 32] = vb2;
  vout[ob + lane] = vb1;    vout[ob + lane + 32] = vb2;
}

// ---------------------------------------------------------------------------
// Flash attention: one wave per 16-row q-tile. Online softmax, WMMA for both
// Q*K^T and P*V. P is re-laid-out C->A through a wave-private LDS scratch
// (intra-wave DS ops are in-order; compiler inserts s_wait_dscnt).
// ---------------------------------------------------------------------------
__global__ __launch_bounds__(128) void flash_attn(const float* __restrict__ qw,
                                                  const float* __restrict__ kw,
                                                  const float* __restrict__ vw,
                                                  float* __restrict__ ao) {
  __shared__ float P[4 * 16 * 18];  // 4 waves x 16x16 tile, pitch 18
  const int wave = threadIdx.x >> 5;
  const int lane = threadIdx.x & 31;
  const int half = lane >> 4;
  const int ln   = lane & 15;
  const int gw   = blockIdx.x * 4 + wave;
  const int bh   = gw >> 7;   // / (L_/16)
  const int qt   = gw & 127;
  const int b    = bh >> 4;
  const int h    = bh & 15;
  const size_t base = (size_t)bh * L_ * HD_;
  float* Pw = &P[wave * (16 * 18)];
  const float scale = 0.125f;  // 1/sqrt(HD)

  // Q tile as 16 A-fragments (K=64), scores scale folded in.
  v2f qA[16];
#pragma unroll
  for (int kf = 0; kf < 16; ++kf) {
    v2f t = *(const v2f*)&qw[base + (size_t)(qt * 16 + ln) * HD_ + kf * 4 + 2 * half];
    qA[kf] = t * scale;
  }

  v8f o[4] = {};
  float m[8], lsum[8];
#pragma unroll
  for (int r = 0; r < 8; ++r) { m[r] = -1e30f; lsum[r] = 0.0f; }

  for (int kt = 0; kt < L_ / 16; ++kt) {
    // S = Q * K^T : B-frag of K^T is contiguous along hd in k_ws.
    v8f sc = {};
#pragma unroll
    for (int kf = 0; kf < 16; ++kf) {
      v2f bB = *(const v2f*)&kw[base + (size_t)(kt * 16 + ln) * HD_ + kf * 4 + 2 * half];
      sc = wmma_f32(qA[kf], bB, sc);
    }

    // Online softmax per row (row M = r + 8*half lives in the 16-lane half).
    float esc[8];
#pragma unroll
    for (int r = 0; r < 8; ++r) {
      const float x = sc[r];
      float mx = x;
#pragma unroll
      for (int off = 1; off < 16; off <<= 1) mx = fmaxf(mx, __shfl_xor(mx, off, 32));
      const float mnew = fmaxf(m[r], mx);
      const float e    = __expf(m[r] - mnew);
      const float p    = __expf(x - mnew);
      float ps = p;
#pragma unroll
      for (int off = 1; off < 16; off <<= 1) ps += __shfl_xor(ps, off, 32);
      lsum[r] = lsum[r] * e + ps;
      m[r]    = mnew;
      esc[r]  = e;
      Pw[(r + 8 * half) * 18 + ln] = p;  // C-layout -> LDS
    }
#pragma unroll
    for (int r = 0; r < 8; ++r) {
      o[0][r] *= esc[r]; o[1][r] *= esc[r]; o[2][r] *= esc[r]; o[3][r] *= esc[r];
    }

    // Reload P in A-layout, then O += P * V.
    v2f pA[4];
#pragma unroll
    for (int kf = 0; kf < 4; ++kf)
      pA[kf] = *(const v2f*)&Pw[ln * 18 + kf * 4 + 2 * half];
#pragma unroll
    for (int nt = 0; nt < 4; ++nt) {
#pragma unroll
      for (int kf = 0; kf < 4; ++kf) {
        const int kr = kt * 16 + kf * 4 + 2 * half;
        v2f bV;
        bV.x = vw[base + (size_t)kr * HD_ + nt * 16 + ln];
        bV.y = vw[base + (size_t)(kr + 1) * HD_ + nt * 16 + ln];
        o[nt] = wmma_f32(pA[kf], bV, o[nt]);
      }
    }
  }

  // Normalize and scatter to (b, l, h*HD + d) layout for the projection GEMM.
#pragma unroll
  for (int r = 0; r < 8; ++r) {
    const float inv = 1.0f / lsum[r];
    const int lq = qt * 16 + r + 8 * half;
    const size_t orow = (size_t)(b * L_ + lq) * DIM_ + h * HD_;
#pragma unroll
    for (int nt = 0; nt < 4; ++nt)
      ao[orow + nt * 16 + ln] = o[nt][r] * inv;
  }
}

// ---------------------------------------------------------------------------
extern "C" void kernel_launch(void* const* d_in, const int* in_sizes, int n_in,
                              void* d_out, int out_size, void* d_ws,
                              size_t ws_size, hipStream_t stream) {
  const float* x     = (const float*)d_in[0];
  const float* v0    = (const float*)d_in[1];
  const float* rc    = (const float*)d_in[2];
  const float* rsn   = (const float*)d_in[3];
  const float* qkvw  = (const float*)d_in[4];
  const float* projw = (const float*)d_in[5];
  const float* lamp  = (const float*)d_in[6];
  float* out = (float*)d_out;

  // Workspace layout (floats): needs ~117.4 MB.
  float* ws      = (float*)d_ws;
  float* qkv_ws  = ws;                        // 4096*3072
  float* q_ws    = qkv_ws + 4096 * 3072;      // 4M
  float* k_ws    = q_ws + 4194304;            // 4M
  float* v_ws    = k_ws + 4194304;            // 4M
  float* attn_ws = v_ws + 4194304;            // 4M

  // 1) QKV GEMM: (4096x1024)*(1024x3072)
  gemm_f32_nt<<<dim3(3072 / 64, 4096 / 128), 256, 0, stream>>>(
      x, qkvw, qkv_ws, 4096, 3072, 1024);

  // 2) blend/RoPE/scale/RMS epilogue; also writes v into d_out tuple slot 2.
  qkv_epilogue<<<(B_ * H_ * L_) / 8, 256, 0, stream>>>(
      qkv_ws, v0, rc, rsn, lamp, q_ws, k_ws, v_ws, out + 4194304);

  // 3) flash attention over all (b,h) heads.
  flash_attn<<<(B_ * H_ * (L_ / 16)) / 4, 128, 0, stream>>>(q_ws, k_ws, v_ws,
                                                            attn_ws);

  // 4) output projection: (4096x1024)*(1024x1024) -> d_out tuple slot 1.
  gemm_f32_nt<<<dim3(1024 / 64, 4096 / 128), 256, 0, stream>>>(
      attn_ws, projw, out, 4096, 1024, 1024);
}